// Encoder_61237643707096
// MI455X (gfx1250) — compile-verified
//
#include <hip/hip_runtime.h>

#define LL   2048
#define BB   512
#define INP  78
#define HID  4
#define GATES 16   // 4*HID

typedef float v2f __attribute__((ext_vector_type(2)));
typedef float v8f __attribute__((ext_vector_type(8)));

// ---------------------------------------------------------------------------
// Fast device math (CDNA5 transcendental VALU ops)
// ---------------------------------------------------------------------------
__device__ __forceinline__ float fexp2(float x) {
#if __has_builtin(__builtin_amdgcn_exp2f)
    return __builtin_amdgcn_exp2f(x);
#else
    return __expf(x * 0.6931471805599453f);
#endif
}
__device__ __forceinline__ float frcp(float x) {
#if __has_builtin(__builtin_amdgcn_rcpf)
    return __builtin_amdgcn_rcpf(x);
#else
    return 1.0f / x;
#endif
}
__device__ __forceinline__ float sigm(float x) {
    // 1/(1+e^-x) = 1/(1+2^(-x*log2e))
    return frcp(1.0f + fexp2(-1.4426950408889634f * x));
}
__device__ __forceinline__ float tanh_(float x) {
#if __has_builtin(__builtin_amdgcn_tanhf)
    return __builtin_amdgcn_tanhf(x);
#else
    float e = fexp2(-2.8853900817779268f * x);   // e^{-2x}
    return (1.0f - e) * frcp(1.0f + e);
#endif
}

// group-of-4 full-sharing ds_swizzle broadcast: every lane in a group of 4
// consecutive lanes receives the value of group-lane K.  0x8000 | K*0x55.
#define BC4(v, MASK) \
    __int_as_float(__builtin_amdgcn_ds_swizzle(__float_as_int(v), (MASK)))

// ---------------------------------------------------------------------------
// Kernel A: xg0[M,16] = x[M,78] @ W_ih0^T[78,16] + (b_ih0 + b_hh0)
// fp32 WMMA 16x16x4, K padded 78 -> 80 (20 WMMAs per tile, 2 acc chains).
// All 20 A-fragments are preloaded so the loads stay in flight together.
// Output columns are permuted: perm(n) = (n&3)*4 + (n>>2) so that the scan
// kernel's per-(batch,unit) lane reads its 4 gate pre-activations as one b128.
// ---------------------------------------------------------------------------
__global__ __launch_bounds__(256)
void xg0_gemm_wmma(const float* __restrict__ x,
                   const float* __restrict__ Wih,   // [16,78]
                   const float* __restrict__ bih,
                   const float* __restrict__ bhh,
                   float* __restrict__ xg)          // [M,16] permuted cols
{
    const int M    = LL * BB;
    const int wave = (int)((blockIdx.x * blockDim.x + threadIdx.x) >> 5);
    const int lane = (int)(threadIdx.x & 31);
    const int m0   = wave * 16;
    if (m0 >= M) return;

    const int half = lane >> 4;    // 0: K=k,k+1  1: K=k+2,k+3
    const int n    = lane & 15;    // output column (gate index)
    const int KCLAMP = INP - 2;    // 76: last fully in-bounds even pair

    // ---- B fragments (W_ih^T), branchless clamp + zero of the one pad pair.
    // B(K,N) = Wih[N*78 + K]; VGPR0 holds K=k0+2*half, VGPR1 holds K=+1.
    const float* wrow = Wih + (size_t)n * INP;
    v2f bfrag[20];
#pragma unroll
    for (int s = 0; s < 20; ++s) {
        int k  = 4 * s + 2 * half;
        int ks = (k > KCLAMP) ? KCLAMP : k;      // only s==19,half==1 clamps
        bfrag[s] = *(const v2f*)(wrow + ks);
    }
    bfrag[19].x = half ? 0.0f : bfrag[19].x;     // K=78,79 -> zero pad
    bfrag[19].y = half ? 0.0f : bfrag[19].y;

    // ---- A fragments: lanes 0-15 / 16-31 both cover rows m0..m0+15; each
    // lane loads an (even,odd) column pair -> 8B-aligned contiguous b64.
    // Issue all 20 loads before any WMMA so they overlap in the memory system.
    const float* xrow = x + (size_t)(m0 + n) * INP;
    v2f afrag[20];
#pragma unroll
    for (int s = 0; s < 20; ++s) {
        int k  = 4 * s + 2 * half;
        int ks = (k > KCLAMP) ? KCLAMP : k;
        afrag[s] = *(const v2f*)(xrow + ks);
    }
    afrag[19].x = half ? 0.0f : afrag[19].x;
    afrag[19].y = half ? 0.0f : afrag[19].y;

    // ---- two independent accumulation chains (halve WMMA RAW depth)
    const float bias = bih[n] + bhh[n];
    v8f acc0, acc1;
#pragma unroll
    for (int r = 0; r < 8; ++r) { acc0[r] = bias; acc1[r] = 0.0f; }

#pragma unroll
    for (int s = 0; s < 20; s += 2) {
        acc0 = __builtin_amdgcn_wmma_f32_16x16x4_f32(
                   false, afrag[s],     false, bfrag[s],     (short)0, acc0,
                   false, false);
        acc1 = __builtin_amdgcn_wmma_f32_16x16x4_f32(
                   false, afrag[s + 1], false, bfrag[s + 1], (short)0, acc1,
                   false, false);
    }
#pragma unroll
    for (int r = 0; r < 8; ++r) acc0[r] += acc1[r];

    // D layout: VGPR r = row m0 + r (lanes 0-15) / m0 + 8 + r (lanes 16-31),
    // column n.  Store with gate permutation for the scan kernel.
    const int perm = (n & 3) * 4 + (n >> 2);
    float* orow = xg + (size_t)(m0 + half * 8) * GATES + perm;
#pragma unroll
    for (int r = 0; r < 8; ++r)
        orow[(size_t)r * GATES] = acc0[r];
}

// ---------------------------------------------------------------------------
// Kernel B: fused 2-layer sequential scan.
// One lane per (batch b, hidden unit u): 2048 lanes = 64 wave32s.
// Hidden vectors (4 wide) are shared inside each 4-lane group via ds_swizzle.
// ---------------------------------------------------------------------------
__global__ __launch_bounds__(32)
void lstm_scan2(const float* __restrict__ xg0,   // [L*B,16] permuted
                const int*   __restrict__ lens,  // [B]
                const float* __restrict__ Whh0,  // [16,4]
                const float* __restrict__ Wih1,  // [16,4]
                const float* __restrict__ Whh1,  // [16,4]
                const float* __restrict__ bih1,
                const float* __restrict__ bhh1,
                float* __restrict__ out)         // [L,B,4]
{
    const int tid = (int)(blockIdx.x * blockDim.x + threadIdx.x); // 0..2047
    const int b   = tid >> 2;
    const int u   = tid & 3;
    if (b >= BB) return;

    // Per-lane weight rows: gate sections i,f,g,o -> rows gsec*4 + u.
    float w0[4][HID], wi1[4][HID], w1[4][HID], b1[4];
#pragma unroll
    for (int gs = 0; gs < 4; ++gs) {
        const int row = gs * 4 + u;
        b1[gs] = bih1[row] + bhh1[row];
#pragma unroll
        for (int k = 0; k < HID; ++k) {
            w0[gs][k]  = Whh0[row * HID + k];
            wi1[gs][k] = Wih1[row * HID + k];
            w1[gs][k]  = Whh1[row * HID + k];
        }
    }

    const int len = lens[b];

    float c0 = 0.0f, c1 = 0.0f, h0own = 0.0f, h1own = 0.0f;
    float h0g[HID] = {0.f, 0.f, 0.f, 0.f};   // broadcast prev-step h0 of batch
    float h1g[HID] = {0.f, 0.f, 0.f, 0.f};   // broadcast prev-step h1 of batch

    const size_t tstride = (size_t)BB * GATES;
    const float* p = xg0 + (size_t)b * GATES + (size_t)u * 4;
    float4 q = *(const float4*)p;            // prefetch t=0

    for (int t = 0; t < LL; ++t) {
        const float4 qc = q;
        // branchless prefetch of step t+1 (clamped re-read of last step)
        const int tn = (t + 1 < LL) ? (t + 1) : (LL - 1);
        q = *(const float4*)(p + (size_t)tn * tstride);

        // ---- layer 0: gates = xg0 + W_hh0 @ h0(prev) ----
        float gi = qc.x, gf = qc.y, gg = qc.z, go = qc.w;
#pragma unroll
        for (int k = 0; k < HID; ++k) {
            gi += w0[0][k] * h0g[k];
            gf += w0[1][k] * h0g[k];
            gg += w0[2][k] * h0g[k];
            go += w0[3][k] * h0g[k];
        }
        c0    = sigm(gf) * c0 + sigm(gi) * tanh_(gg);
        h0own = sigm(go) * tanh_(c0);

        // share new h0 within the 4-lane batch group
        h0g[0] = BC4(h0own, 0x8000);
        h0g[1] = BC4(h0own, 0x8055);
        h0g[2] = BC4(h0own, 0x80AA);
        h0g[3] = BC4(h0own, 0x80FF);

        // ---- layer 1: gates = b1 + W_ih1 @ h0(new) + W_hh1 @ h1(prev) ----
        float Gi = b1[0], Gf = b1[1], Gg = b1[2], Go = b1[3];
#pragma unroll
        for (int k = 0; k < HID; ++k) {
            Gi += wi1[0][k] * h0g[k] + w1[0][k] * h1g[k];
            Gf += wi1[1][k] * h0g[k] + w1[1][k] * h1g[k];
            Gg += wi1[2][k] * h0g[k] + w1[2][k] * h1g[k];
            Go += wi1[3][k] * h0g[k] + w1[3][k] * h1g[k];
        }
        c1    = sigm(Gf) * c1 + sigm(Gi) * tanh_(Gg);
        h1own = sigm(Go) * tanh_(c1);

        // share new h1 for next step's recurrence
        h1g[0] = BC4(h1own, 0x8000);
        h1g[1] = BC4(h1own, 0x8055);
        h1g[2] = BC4(h1own, 0x80AA);
        h1g[3] = BC4(h1own, 0x80FF);

        // masked output (reference zeroes t >= len); contiguous b32 per lane
        out[(size_t)t * (BB * HID) + tid] = (t < len) ? h1own : 0.0f;
    }
}

// ---------------------------------------------------------------------------
extern "C" void kernel_launch(void* const* d_in, const int* in_sizes, int n_in,
                              void* d_out, int out_size, void* d_ws, size_t ws_size,
                              hipStream_t stream)
{
    const float* x    = (const float*)d_in[0];
    const int*   lens = (const int*)  d_in[1];
    const float* Wih0 = (const float*)d_in[2];
    const float* Whh0 = (const float*)d_in[3];
    const float* bih0 = (const float*)d_in[4];
    const float* bhh0 = (const float*)d_in[5];
    const float* Wih1 = (const float*)d_in[6];
    const float* Whh1 = (const float*)d_in[7];
    const float* bih1 = (const float*)d_in[8];
    const float* bhh1 = (const float*)d_in[9];
    float* out = (float*)d_out;
    float* xg0 = (float*)d_ws;   // L*B*16 floats = 64 MB scratch

    // Phase 1: WMMA fp32 input projection.  65536 16x16 tiles, 8 waves/block.
    const int tiles  = (LL * BB) / 16;
    const int blocks = tiles / 8;
    hipLaunchKernelGGL(xg0_gemm_wmma, dim3(blocks), dim3(256), 0, stream,
                       x, Wih0, bih0, bhh0, xg0);

    // Phase 2: fused 2-layer scan.  2048 lanes -> 64 wave32s, one per block.
    hipLaunchKernelGGL(lstm_scan2, dim3((BB * HID) / 32), dim3(32), 0, stream,
                       xg0, lens, Whh0, Wih1, Whh1, bih1, bhh1, out);
}